// Net_15908558864408
// MI455X (gfx1250) — compile-verified
//
#include <hip/hip_runtime.h>

// SNN forward: x[10,B,4] -> (spk2[10,B,3], mem2[10,B,3]), B = 1,000,000.
// Memory-bound floor ~17 us (400 MB @ 23.3 TB/s); VALU was the marginal
// bottleneck at 16 elements/wave, so this version packs 32 elements/wave:
// two V_WMMA_F32_16X16X4_F32 per step with chained accumulator:
//   D = A2(w1 in rows 8..12) x B2(elems 16..31)  +  A1(w1 in rows 0..4) x B1(elems 0..15)
// Disjoint zero rows make the merge exact; lane L ends up owning element
// base+L with cur1[0..4] in d[0..4]. LIF recurrence + 5x3 layer-2 matvec run
// at full 32-lane utilization in registers.

typedef __attribute__((ext_vector_type(2))) float v2f;
typedef __attribute__((ext_vector_type(8))) float v8f;

#define NUM_STEPS  10
#define NUM_INPUT   4
#define NUM_HIDDEN  5
#define NUM_OUTPUT  3

__global__ __launch_bounds__(256) void snn_lif_wmma_kernel(
    const float* __restrict__ x,    // [NUM_STEPS, batch, NUM_INPUT]
    const float* __restrict__ w1,   // [NUM_HIDDEN, NUM_INPUT]
    const float* __restrict__ w2,   // [NUM_OUTPUT, NUM_HIDDEN]
    float* __restrict__ out_spk,    // [NUM_STEPS, batch, NUM_OUTPUT]
    float* __restrict__ out_mem,    // [NUM_STEPS, batch, NUM_OUTPUT]
    int batch)
{
    const float BETA = 0.95f;
    const float TH   = 1.0f;

    const int lane   = threadIdx.x & 31;
    const int waveId = blockIdx.x * (blockDim.x >> 5) + (threadIdx.x >> 5);
    const long base  = (long)waveId * 32;        // 32 elements per wave
    if (base >= batch) return;                   // wave-uniform exit

    const int row = lane & 15;        // A-row / B-column index this lane serves
    const int kp  = (lane >> 4) << 1; // K-pair this half-wave supplies: 0 or 2

    // ---- A1: w1 (5x4) in rows 0..4 of 16x4; A2: w1 in rows 8..12.
    v2f a1, a2;
    a1.x = (row < NUM_HIDDEN) ? w1[row * NUM_INPUT + kp]     : 0.0f;
    a1.y = (row < NUM_HIDDEN) ? w1[row * NUM_INPUT + kp + 1] : 0.0f;
    const int r2 = row - 8;
    const bool v2 = (r2 >= 0) && (r2 < NUM_HIDDEN);
    a2.x = v2 ? w1[r2 * NUM_INPUT + kp]     : 0.0f;
    a2.y = v2 ? w1[r2 * NUM_INPUT + kp + 1] : 0.0f;

    // ---- w2 broadcast (uniform -> scalar loads)
    float w2r[NUM_OUTPUT][NUM_HIDDEN];
    #pragma unroll
    for (int o = 0; o < NUM_OUTPUT; ++o)
        #pragma unroll
        for (int h = 0; h < NUM_HIDDEN; ++h)
            w2r[o][h] = w2[o * NUM_HIDDEN + h];

    float mem1[NUM_HIDDEN] = {0.f, 0.f, 0.f, 0.f, 0.f};
    float mem2[NUM_OUTPUT] = {0.f, 0.f, 0.f};

    // ---- element this lane owns (lanes 0-15 -> group 0, 16-31 -> group 1)
    const long elem = base + lane;

    // ---- B-tile column indices (clamped so partial tail tiles load safely;
    //      stores are guarded per-lane below)
    long e0 = base + row;        if (e0 >= batch) e0 = batch - 1;  // group 0 cols
    long e1 = base + 16 + row;   if (e1 >= batch) e1 = batch - 1;  // group 1 cols

    // B layout (4x16): lane n holds K rows {kp, kp+1}. Wave reads 2x256B
    // contiguous blocks of x per step.
    v2f b0 = *(const v2f*)(x + ((long)0 * batch + e0) * NUM_INPUT + kp);
    v2f b1 = *(const v2f*)(x + ((long)0 * batch + e1) * NUM_INPUT + kp);

    #pragma unroll
    for (int t = 0; t < NUM_STEPS; ++t) {
        v2f b0n = b0, b1n = b1;
        if (t + 1 < NUM_STEPS) {
            b0n = *(const v2f*)(x + ((long)(t + 1) * batch + e0) * NUM_INPUT + kp);
            b1n = *(const v2f*)(x + ((long)(t + 1) * batch + e1) * NUM_INPUT + kp);
        }
        if (t + 2 < NUM_STEPS) {
            __builtin_prefetch(x + ((long)(t + 2) * batch + e0) * NUM_INPUT + kp, 0, 1);
            __builtin_prefetch(x + ((long)(t + 2) * batch + e1) * NUM_INPUT + kp, 0, 1);
        }

        // cur1 for 32 elements via two chained WMMAs (disjoint nonzero rows)
        v8f c = {};
        v8f d = __builtin_amdgcn_wmma_f32_16x16x4_f32(
            false, a1, false, b0, (short)0, c, false, false);
        d = __builtin_amdgcn_wmma_f32_16x16x4_f32(
            false, a2, false, b1, (short)0, d, false, false);
        // Every lane: cur1[h] = d[h], h = 0..4, for element `elem`.

        // ---- Layer 1 LIF (full 32-lane utilization)
        float spk1[NUM_HIDDEN];
        #pragma unroll
        for (int h = 0; h < NUM_HIDDEN; ++h) {
            float reset = ((mem1[h] - TH) > 0.0f) ? 1.0f : 0.0f;  // pre-update mem
            mem1[h] = BETA * mem1[h] + d[h] - reset * TH;
            spk1[h] = ((mem1[h] - TH) > 0.0f) ? 1.0f : 0.0f;
        }

        // ---- Layer 2: 5x3 matvec + LIF in registers
        float spk2[NUM_OUTPUT];
        #pragma unroll
        for (int o = 0; o < NUM_OUTPUT; ++o) {
            float cur2 = 0.0f;
            #pragma unroll
            for (int h = 0; h < NUM_HIDDEN; ++h)
                cur2 = fmaf(w2r[o][h], spk1[h], cur2);
            float reset = ((mem2[o] - TH) > 0.0f) ? 1.0f : 0.0f;
            mem2[o] = BETA * mem2[o] + cur2 - reset * TH;
            spk2[o] = ((mem2[o] - TH) > 0.0f) ? 1.0f : 0.0f;
        }

        // ---- Record: full-wave 384B contiguous stores per array
        if (elem < batch) {
            long ob = ((long)t * batch + elem) * NUM_OUTPUT;
            out_spk[ob + 0] = spk2[0];
            out_spk[ob + 1] = spk2[1];
            out_spk[ob + 2] = spk2[2];
            out_mem[ob + 0] = mem2[0];
            out_mem[ob + 1] = mem2[1];
            out_mem[ob + 2] = mem2[2];
        }

        b0 = b0n;
        b1 = b1n;
    }
}

extern "C" void kernel_launch(void* const* d_in, const int* in_sizes, int n_in,
                              void* d_out, int out_size, void* d_ws, size_t ws_size,
                              hipStream_t stream) {
    const float* x  = (const float*)d_in[0];   // [10, B, 4]
    const float* w1 = (const float*)d_in[1];   // [5, 4]
    const float* w2 = (const float*)d_in[2];   // [3, 5]

    const int batch = in_sizes[0] / (NUM_STEPS * NUM_INPUT);

    float* out_spk = (float*)d_out;                                     // [10, B, 3]
    float* out_mem = out_spk + (size_t)NUM_STEPS * batch * NUM_OUTPUT;  // [10, B, 3]

    const int threads = 256;                        // 8 waves/block
    const int elemsPerBlock = (threads / 32) * 32;  // 256 elements/block
    const int grid = (batch + elemsPerBlock - 1) / elemsPerBlock;

    snn_lif_wmma_kernel<<<grid, threads, 0, stream>>>(x, w1, w2, out_spk, out_mem, batch);
}